// StructuralFingerprintLayer_25022479466559
// MI455X (gfx1250) — compile-verified
//
#include <hip/hip_runtime.h>
#include <math.h>
#include <stdint.h>

typedef __attribute__((ext_vector_type(2))) float v2f;
typedef __attribute__((ext_vector_type(8))) float v8f;

#define NN    8192
#define FIN   512
#define FOUT  64
#define ALPHA 0.2f

// pass2 tiling
#define TJ     128            // j-columns per tile
#define TJS    132            // padded LDS row stride (floats): banks 4*m, conflict-free
#define BUFB   (16 * TJS * 4) // bytes per 16-row tile buffer = 8448
#define NTILES (NN / TJ)      // 64

// ---------------------------------------------------------------------------
// CDNA5 async global->LDS copy (ASYNCcnt) — inline asm, no builtin exists.
// lds_off is a byte offset from the workgroup LDS base (single __shared__
// block declared at offset 0).
// ---------------------------------------------------------------------------
__device__ __forceinline__ void async_load_b128(uint32_t lds_off, const void* gptr) {
    uint64_t ga = (uint64_t)(uintptr_t)gptr;
    asm volatile("global_load_async_to_lds_b128 %0, %1, off"
                 :: "v"(lds_off), "v"(ga) : "memory");
}
__device__ __forceinline__ void wait_async0() {
    asm volatile("s_wait_asynccnt 0" ::: "memory");
}

// ---------------------------------------------------------------------------
// Kernel 1: seq_fts = x @ W.T   (fp32 WMMA 16x16x4, one wave per 16x16 tile)
// ---------------------------------------------------------------------------
__global__ __launch_bounds__(32)
void proj_wmma_kernel(const float* __restrict__ x, const float* __restrict__ W,
                      float* __restrict__ seq) {
    const int i0 = blockIdx.x * 16;
    const int c0 = blockIdx.y * 16;
    const int l  = threadIdx.x;
    const int m  = l & 15;
    const int hi = l >> 4;

    v8f acc = {};
    const float* ap = x + (size_t)(i0 + m) * FIN + 2 * hi;
    const float* bp = W + (size_t)(c0 + m) * FIN + 2 * hi;  // B[k][n] = W[n][k]
    for (int k0 = 0; k0 < FIN; k0 += 4) {
        v2f a; a.x = ap[k0];  a.y = ap[k0 + 1];
        v2f b; b.x = bp[k0];  b.y = bp[k0 + 1];
        acc = __builtin_amdgcn_wmma_f32_16x16x4_f32(
            false, a, false, b, (short)0, acc, false, false);
    }
#pragma unroll
    for (int r = 0; r < 8; ++r)
        seq[(size_t)(i0 + r + 8 * hi) * FOUT + c0 + m] = acc[r];
}

// ---------------------------------------------------------------------------
// Kernel 2: f1 = seq_fts @ a1 + b1 ; f2 = seq_fts @ a2 + b2
// ---------------------------------------------------------------------------
__global__ __launch_bounds__(256)
void f12_kernel(const float* __restrict__ seq,
                const float* __restrict__ a1, const float* __restrict__ b1,
                const float* __restrict__ a2, const float* __restrict__ b2,
                float* __restrict__ f1, float* __restrict__ f2) {
    int i = blockIdx.x * blockDim.x + threadIdx.x;
    if (i >= NN) return;
    const float* r = seq + (size_t)i * FOUT;
    float s1 = 0.f, s2 = 0.f;
#pragma unroll 8
    for (int k = 0; k < FOUT; ++k) {
        float v = r[k];
        s1 = fmaf(v, a1[k], s1);
        s2 = fmaf(v, a2[k], s2);
    }
    f1[i] = s1 + b1[0];
    f2[i] = s2 + b2[0];
}

// ---------------------------------------------------------------------------
// Kernel 3: first softmax row stats: M1[i], invS1[i]  (reads adj once)
// ---------------------------------------------------------------------------
__global__ __launch_bounds__(256)
void pass1_kernel(const float* __restrict__ adj, const float* __restrict__ f1,
                  const float* __restrict__ f2, float* __restrict__ M1,
                  float* __restrict__ invS1) {
    const int i = blockIdx.x;
    const float fi = f1[i];
    const float* row = adj + (size_t)i * NN;

    float lm = -3.4e38f, ls = 0.f;
    for (int j = threadIdx.x; j < NN; j += 256) {
        float t = fi + f2[j];
        t = (t > 0.f) ? t : ALPHA * t;
        t += row[j];
        if (t > lm) { ls = ls * __expf(lm - t) + 1.f; lm = t; }
        else        { ls += __expf(t - lm); }
    }
    __shared__ float sm[256], ss[256];
    sm[threadIdx.x] = lm; ss[threadIdx.x] = ls;
    __syncthreads();
    for (int off = 128; off >= 1; off >>= 1) {
        if (threadIdx.x < off) {
            float mA = sm[threadIdx.x], sA = ss[threadIdx.x];
            float mB = sm[threadIdx.x + off], sB = ss[threadIdx.x + off];
            float M = fmaxf(mA, mB);
            sm[threadIdx.x] = M;
            ss[threadIdx.x] = sA * __expf(mA - M) + sB * __expf(mB - M);
        }
        __syncthreads();
    }
    if (threadIdx.x == 0) { M1[i] = sm[0]; invS1[i] = 1.f / ss[0]; }
}

// ---------------------------------------------------------------------------
// Kernel 4: fused second softmax + coefs@seq_fts (WMMA f32) + ELU.
// adj/adj_ad streamed via double-buffered GLOBAL_LOAD_ASYNC_TO_LDS_B128.
// One workgroup (8 waves) per 16-row tile; waves partition each j-tile.
// ---------------------------------------------------------------------------
__global__ __launch_bounds__(256)
void pass2_wmma_kernel(const float* __restrict__ adj,
                       const float* __restrict__ adj_ad,
                       const float* __restrict__ seq,
                       const float* __restrict__ f1, const float* __restrict__ f2,
                       const float* __restrict__ M1, const float* __restrict__ invS1,
                       const float* __restrict__ bias,
                       const float* __restrict__ w_ei, const float* __restrict__ w_si,
                       float* __restrict__ out) {
    // Single LDS block => known base offset 0 for the async-load asm.
    // [0, 2*BUFB)            : adj tiles, 2 buffers
    // [2*BUFB, 4*BUFB)       : adj_ad tiles, 2 buffers
    // [4*BUFB, +32KB)        : hacc[8][16][64]
    // then sacc[8][16]
    __shared__ __align__(16) unsigned char smem[4 * BUFB + 32768 + 512];
    float* adjT = (float*)smem;
    float* adT  = (float*)(smem + 2 * BUFB);
    float (*hacc)[16][FOUT] = (float (*)[16][FOUT])(smem + 4 * BUFB);
    float (*sacc)[16]       = (float (*)[16])(smem + 4 * BUFB + 32768);

    const int tid = threadIdx.x;
    const int i0  = blockIdx.x * 16;
    const int l   = tid & 31;
    const int w   = tid >> 5;     // wave 0..7
    const int m   = l & 15;
    const int hi  = l >> 4;

    // loader mapping: row lr, 16B segment ls
    const int lr = tid >> 4;      // 0..15
    const int ls = tid & 15;      // 0..15

    const float aw = fabsf(w_ei[0]);
    const float sw = fabsf(w_si[0]);
    const float shift = aw;       // constant shift: softmax shift-invariant

    const int   i   = i0 + m;
    const float fi  = f1[i];
    const float m1i = M1[i];
    const float is1 = invS1[i];

    v8f acc0 = {}, acc1 = {}, acc2 = {}, acc3 = {};
    float s2local = 0.f;

    const size_t growA = (size_t)(i0 + lr) * NN;  // loader's adj row base

    // prime tile 0
    {
        const int buf = 0;
#pragma unroll
        for (int q = 0; q < 2; ++q) {
            const int col = ls * 4 + q * 64;
            async_load_b128((uint32_t)(buf * BUFB + (lr * TJS + col) * 4),
                            adj + growA + col);
            async_load_b128((uint32_t)(2 * BUFB + buf * BUFB + (lr * TJS + col) * 4),
                            adj_ad + growA + col);
        }
    }
    wait_async0();
    __syncthreads();

    for (int t = 0; t < NTILES; ++t) {
        // issue async loads for next tile into the other buffer
        if (t + 1 < NTILES) {
            const int buf = (t + 1) & 1;
            const size_t g = growA + (size_t)(t + 1) * TJ;
#pragma unroll
            for (int q = 0; q < 2; ++q) {
                const int col = ls * 4 + q * 64;
                async_load_b128((uint32_t)(buf * BUFB + (lr * TJS + col) * 4),
                                adj + g + col);
                async_load_b128((uint32_t)(2 * BUFB + buf * BUFB + (lr * TJS + col) * 4),
                                adj_ad + g + col);
            }
        }

        // compute on current tile from LDS
        const float* at = adjT + (t & 1) * (16 * TJS);
        const float* dt = adT  + (t & 1) * (16 * TJS);
#pragma unroll
        for (int s = 0; s < TJ / 32; ++s) {       // 4 chunks of 4 j per wave
            const int jloc = w * 4 + s * 32;
            const int jl   = jloc + 2 * hi;       // lane's K columns {jl, jl+1}
            const int j    = t * TJ + jl;

            float2 av = *(const float2*)&at[m * TJS + jl];  // ds_load_b64
            float2 dv = *(const float2*)&dt[m * TJS + jl];
            float2 fj = *(const float2*)(f2 + j);

            float t0 = fi + fj.x; t0 = (t0 > 0.f) ? t0 : ALPHA * t0; t0 += av.x;
            float t1 = fi + fj.y; t1 = (t1 > 0.f) ? t1 : ALPHA * t1; t1 += av.y;
            float e0 = __expf(t0 - m1i) * is1;
            float e1 = __expf(t1 - m1i) * is1;
            float c0 = __expf(fmaf(aw, e0, sw * dv.x) - shift);
            float c1 = __expf(fmaf(aw, e1, sw * dv.y) - shift);
            s2local += c0 + c1;

            v2f a; a.x = c0; a.y = c1;            // A[M=m][K=2*hi+v]

            const float* b0p = seq + (size_t)(t * TJ + jloc + 2 * hi) * FOUT + m;
            const float* b1p = b0p + FOUT;
            v2f b;
            b.x = b0p[0];  b.y = b1p[0];
            acc0 = __builtin_amdgcn_wmma_f32_16x16x4_f32(false, a, false, b, (short)0, acc0, false, false);
            b.x = b0p[16]; b.y = b1p[16];
            acc1 = __builtin_amdgcn_wmma_f32_16x16x4_f32(false, a, false, b, (short)0, acc1, false, false);
            b.x = b0p[32]; b.y = b1p[32];
            acc2 = __builtin_amdgcn_wmma_f32_16x16x4_f32(false, a, false, b, (short)0, acc2, false, false);
            b.x = b0p[48]; b.y = b1p[48];
            acc3 = __builtin_amdgcn_wmma_f32_16x16x4_f32(false, a, false, b, (short)0, acc3, false, false);
        }

        // next-tile data landed + all waves done reading current buffer
        wait_async0();
        __syncthreads();
    }

    // combine K-halves of the row sum (lanes m and m+16 both own row m)
    float s2row = s2local + __shfl_xor(s2local, 16, 32);
    if (hi == 0) sacc[w][m] = s2row;

    // dump accumulators: vgpr r, lane l -> (row r + 8*hi, col n*16 + m)
#pragma unroll
    for (int r = 0; r < 8; ++r) {
        hacc[w][r + 8 * hi][ 0 + m] = acc0[r];
        hacc[w][r + 8 * hi][16 + m] = acc1[r];
        hacc[w][r + 8 * hi][32 + m] = acc2[r];
        hacc[w][r + 8 * hi][48 + m] = acc3[r];
    }
    __syncthreads();

    // final reduce across 8 waves, normalize, bias, ELU, store
    const int row = tid >> 4;
    const int cb  = (tid & 15) * 4;
    float s2 = 0.f;
#pragma unroll
    for (int ww = 0; ww < 8; ++ww) s2 += sacc[ww][row];
    const float inv = 1.f / s2;
#pragma unroll
    for (int c = 0; c < 4; ++c) {
        float h = 0.f;
#pragma unroll
        for (int ww = 0; ww < 8; ++ww) h += hacc[ww][row][cb + c];
        h = fmaf(h, inv, bias[cb + c]);
        out[(size_t)(i0 + row) * FOUT + cb + c] = (h > 0.f) ? h : (__expf(h) - 1.f);
    }
}

// ---------------------------------------------------------------------------
extern "C" void kernel_launch(void* const* d_in, const int* in_sizes, int n_in,
                              void* d_out, int out_size, void* d_ws, size_t ws_size,
                              hipStream_t stream) {
    const float* x      = (const float*)d_in[0];
    const float* adj    = (const float*)d_in[1];
    const float* adj_ad = (const float*)d_in[2];
    const float* W      = (const float*)d_in[3];
    const float* a1     = (const float*)d_in[4];
    const float* b1     = (const float*)d_in[5];
    const float* a2     = (const float*)d_in[6];
    const float* b2     = (const float*)d_in[7];
    const float* bias   = (const float*)d_in[8];
    const float* w_ei   = (const float*)d_in[9];
    const float* w_si   = (const float*)d_in[10];
    float* out = (float*)d_out;

    float* ws  = (float*)d_ws;
    float* seq = ws;                       // 8192*64
    float* f1  = seq + (size_t)NN * FOUT;  // 8192
    float* f2  = f1 + NN;                  // 8192
    float* M1  = f2 + NN;                  // 8192
    float* iS1 = M1 + NN;                  // 8192

    proj_wmma_kernel<<<dim3(NN / 16, FOUT / 16), 32, 0, stream>>>(x, W, seq);
    f12_kernel<<<NN / 256, 256, 0, stream>>>(seq, a1, b1, a2, b2, f1, f2);
    pass1_kernel<<<NN, 256, 0, stream>>>(adj, f1, f2, M1, iS1);
    pass2_wmma_kernel<<<NN / 16, 256, 0, stream>>>(adj, adj_ad, seq, f1, f2,
                                                   M1, iS1, bias, w_ei, w_si, out);
}